// GAT_62053687493149
// MI455X (gfx1250) — compile-verified
//
#include <hip/hip_runtime.h>
#include <math.h>

// ---------------------------------------------------------------------------
// 2-layer GAT for MI455X (gfx1250, wave32).
// Dense transforms: V_WMMA_F32_16X16X4_F32 (fp32 tensor path, matches the
// fp32 reference precision). A-tiles staged to LDS via
// GLOBAL_LOAD_ASYNC_TO_LDS_B128 (ASYNCcnt-tracked DMA) when available.
// Edge softmax/aggregation is HBM/atomic bound: float atomicMax/atomicAdd
// (global_atomic_max_num_f32 / global_atomic_add_f32).
// ---------------------------------------------------------------------------

typedef __attribute__((ext_vector_type(2))) float v2f;
typedef __attribute__((ext_vector_type(8))) float v8f;
typedef int v4i __attribute__((vector_size(16)));  // matches builtin param

#define GAT_AS1 __attribute__((address_space(1)))
#define GAT_AS3 __attribute__((address_space(3)))

#if __has_builtin(__builtin_amdgcn_global_load_async_to_lds_b128) && \
    __has_builtin(__builtin_amdgcn_s_wait_asynccnt)
#define GAT_ASYNC_LDS 1
#else
#define GAT_ASYNC_LDS 0
#endif

// C[M,NCOLS] = A[M,KT] @ B[KT,NCOLS]
// One wave -> one 16x16 C tile; WAVES waves share one 16-row A strip in LDS
// (pad +4 floats: frag read addr = sub*(KT+4) + k + 2*half -> banks
// {4*sub+2*half+j} cover 0..63 exactly once; conflict-free b64 reads).
//
// WMMA fragment layout (ISA 7.12.2):
//  A (16x4 f32): lane = half*16+sub holds a[j] = A[M=sub, K=k+2*half+j]
//  B (4x16 f32): lane = half*16+sub holds b[j] = B[K=k+2*half+j, N=c0+sub]
//  C/D: vgpr i, lane half/sub -> C[M = i + 8*half, N = sub]
template <int KT, int NCOLS, int WAVES>
__global__ __launch_bounds__(WAVES * 32) void gemm_f32_wmma(
    const float* __restrict__ A, const float* __restrict__ B,
    float* __restrict__ C, int M) {
  constexpr int LDA = KT + 4;  // (KT+4)*4 bytes per row: 16B-aligned rows
  __shared__ __align__(16) float As[16 * LDA];
  const int tid = threadIdx.x;
  const int wave = tid >> 5;
  const int lane = tid & 31;
  const int half = lane >> 4;
  const int sub = lane & 15;
  const int r0 = blockIdx.x * 16;

#if GAT_ASYNC_LDS
  {  // Async DMA: 16-byte chunks, ASYNCcnt-tracked, no VGPR staging.
    constexpr int CHUNKS = 16 * KT / 4;
    for (int idx = tid; idx < CHUNKS; idx += WAVES * 32) {
      int row = idx / (KT / 4);
      int c4 = idx - row * (KT / 4);
      int gr = r0 + row;
      gr = gr < M ? gr : (M - 1);
      const float* gsrc = A + (size_t)gr * KT + c4 * 4;
      float* ldst = &As[row * LDA + c4 * 4];
      __builtin_amdgcn_global_load_async_to_lds_b128(
          (GAT_AS1 v4i*)gsrc, (GAT_AS3 v4i*)ldst, 0, 0);
    }
    __builtin_amdgcn_s_wait_asynccnt(0);
  }
#else
  for (int idx = tid; idx < 16 * KT; idx += WAVES * 32) {
    int row = idx / KT;
    int col = idx - row * KT;
    int gr = r0 + row;
    gr = gr < M ? gr : (M - 1);
    As[row * LDA + col] = A[(size_t)gr * KT + col];
  }
#endif
  __syncthreads();

  const int c0 = (blockIdx.y * WAVES + wave) * 16;
  if (c0 >= NCOLS) return;  // wave-uniform: EXEC stays all-ones for WMMA

  v8f acc = {};
  const float* Bp = B + (size_t)(2 * half) * NCOLS + c0 + sub;
  const float* Ap = As + sub * LDA + 2 * half;
#pragma unroll 8
  for (int k = 0; k < KT; k += 4) {
    v2f a, b;
    a.x = Ap[k + 0];
    a.y = Ap[k + 1];
    b.x = Bp[(size_t)k * NCOLS];           // imm offset: k*NCOLS*4 bytes
    b.y = Bp[(size_t)(k + 1) * NCOLS];
    // 8 args: (neg_a, A, neg_b, B, c_mod, C, reuse_a, reuse_b)
    acc = __builtin_amdgcn_wmma_f32_16x16x4_f32(false, a, false, b, (short)0,
                                                acc, false, false);
  }

  if (r0 + 16 <= M) {  // block-uniform fast path: imm-offset store clause
    float* Cp = C + (size_t)(r0 + 8 * half) * NCOLS + c0 + sub;
#pragma unroll
    for (int i = 0; i < 8; ++i) Cp[(size_t)i * NCOLS] = acc[i];
  } else {
#pragma unroll
    for (int i = 0; i < 8; ++i) {
      int row = r0 + i + 8 * half;
      if (row < M) C[(size_t)row * NCOLS + c0 + sub] = acc[i];
    }
  }
}

// p[t] = v
__global__ void fill_f32(float* __restrict__ p, long long n, float v) {
  long long t = (long long)blockIdx.x * blockDim.x + threadIdx.x;
  if (t < n) p[t] = v;
}

// a_src[n,h] = sum_c h[n,h,c]*att_src[h,c]; same for a_dst.
template <int H, int C>
__global__ void node_att(const float* __restrict__ h,
                         const float* __restrict__ att_src,
                         const float* __restrict__ att_dst,
                         float* __restrict__ a_src, float* __restrict__ a_dst,
                         int N) {
  int t = blockIdx.x * blockDim.x + threadIdx.x;
  if (t >= N * H) return;
  int hh = t % H;
  const float* hp = h + (size_t)t * C;
  float s = 0.f, d = 0.f;
#pragma unroll
  for (int c = 0; c < C; ++c) {
    float v = hp[c];
    s += v * att_src[hh * C + c];
    d += v * att_dst[hh * C + c];
  }
  a_src[t] = s;
  a_dst[t] = d;
}

__device__ __forceinline__ void edge_sd(int e, int E, const int* __restrict__ src,
                                        const int* __restrict__ dst, int& s,
                                        int& d) {
  if (e < E) {
    s = src[e];
    d = dst[e];
  } else {  // self-loops appended after real edges
    s = d = e - E;
  }
}

__device__ __forceinline__ float lrelu(float x) {
  return x > 0.f ? x : 0.2f * x;
}

// Pass 1: segment max of leaky_relu(a_src[s]+a_dst[d]) over incoming edges.
template <int H>
__global__ void edge_max(const int* __restrict__ src, const int* __restrict__ dst,
                         const float* __restrict__ a_src,
                         const float* __restrict__ a_dst, float* __restrict__ m,
                         int E, int N) {
  long long t = (long long)blockIdx.x * blockDim.x + threadIdx.x;
  long long e = t / H;
  int h = (int)(t - e * H);
  if (e >= (long long)E + N) return;
  int s, d;
  edge_sd((int)e, E, src, dst, s, d);
  float l = lrelu(a_src[s * H + h] + a_dst[d * H + h]);
  atomicMax(&m[d * H + h], l);
}

// Pass 2: segment sum of exp(l - m[dst]).
template <int H>
__global__ void edge_expsum(const int* __restrict__ src,
                            const int* __restrict__ dst,
                            const float* __restrict__ a_src,
                            const float* __restrict__ a_dst,
                            const float* __restrict__ m, float* __restrict__ ssum,
                            int E, int N) {
  long long t = (long long)blockIdx.x * blockDim.x + threadIdx.x;
  long long e = t / H;
  int h = (int)(t - e * H);
  if (e >= (long long)E + N) return;
  int s, d;
  edge_sd((int)e, E, src, dst, s, d);
  float l = lrelu(a_src[s * H + h] + a_dst[d * H + h]);
  atomicAdd(&ssum[d * H + h], __expf(l - m[d * H + h]));
}

// Pass 3: agg[dst, h, c] += h[src, h, c] * alpha.  Thread per (edge, channel).
template <int H, int C>
__global__ void edge_agg(const int* __restrict__ src, const int* __restrict__ dst,
                         const float* __restrict__ a_src,
                         const float* __restrict__ a_dst,
                         const float* __restrict__ m,
                         const float* __restrict__ ssum,
                         const float* __restrict__ hfeat,
                         float* __restrict__ agg, int E, int N) {
  constexpr int HC = H * C;
  long long t = (long long)blockIdx.x * blockDim.x + threadIdx.x;
  long long e = t / HC;
  int c = (int)(t - e * HC);
  if (e >= (long long)E + N) return;
  int s, d;
  edge_sd((int)e, E, src, dst, s, d);
  int hh = c / C;
  float l = lrelu(a_src[s * H + hh] + a_dst[d * H + hh]);
  float ex = __expf(l - m[d * H + hh]);
  float alpha = ex / (ssum[d * H + hh] + 1e-16f);
  atomicAdd(&agg[(size_t)d * HC + c], hfeat[(size_t)s * HC + c] * alpha);
}

// out = elu(agg + bias)  (in place)
__global__ void bias_elu(float* __restrict__ agg, const float* __restrict__ bias,
                         long long total, int HC) {
  long long t = (long long)blockIdx.x * blockDim.x + threadIdx.x;
  if (t >= total) return;
  float v = agg[t] + bias[(int)(t % HC)];
  agg[t] = v > 0.f ? v : (__expf(v) - 1.f);
}

// out[n,:] = log_softmax(agg[n,:] + bias) over 16 channels.
__global__ void bias_logsoftmax16(const float* __restrict__ agg,
                                  const float* __restrict__ bias,
                                  float* __restrict__ out, int N) {
  int n = blockIdx.x * blockDim.x + threadIdx.x;
  if (n >= N) return;
  float v[16];
  float mx = -INFINITY;
#pragma unroll
  for (int c = 0; c < 16; ++c) {
    v[c] = agg[(size_t)n * 16 + c] + bias[c];
    mx = fmaxf(mx, v[c]);
  }
  float s = 0.f;
#pragma unroll
  for (int c = 0; c < 16; ++c) s += __expf(v[c] - mx);
  float ls = __logf(s);
#pragma unroll
  for (int c = 0; c < 16; ++c) out[(size_t)n * 16 + c] = v[c] - mx - ls;
}

static inline int cdiv(long long a, long long b) {
  return (int)((a + b - 1) / b);
}

extern "C" void kernel_launch(void* const* d_in, const int* in_sizes, int n_in,
                              void* d_out, int out_size, void* d_ws,
                              size_t ws_size, hipStream_t stream) {
  const float* x = (const float*)d_in[0];
  const int* ei = (const int*)d_in[1];
  const float* W1 = (const float*)d_in[2];
  const float* attS1 = (const float*)d_in[3];
  const float* attD1 = (const float*)d_in[4];
  const float* b1 = (const float*)d_in[5];
  const float* W2 = (const float*)d_in[6];
  const float* attS2 = (const float*)d_in[7];
  const float* attD2 = (const float*)d_in[8];
  const float* b2 = (const float*)d_in[9];
  float* out = (float*)d_out;

  const int D = 256;
  const int N = in_sizes[0] / D;
  const int E = in_sizes[1] / 2;
  const long long ET = (long long)E + N;
  const int* srcI = ei;       // edge_index[0]
  const int* dstI = ei + E;   // edge_index[1]

  // Workspace carve-up (floats); ~N*324 floats total (~130 MB for N=100k).
  float* p = (float*)d_ws;
  float* h1 = p;    p += (size_t)N * 128;
  float* aS1 = p;   p += (size_t)N * 8;
  float* aD1 = p;   p += (size_t)N * 8;
  float* m1 = p;    p += (size_t)N * 8;
  float* s1 = p;    p += (size_t)N * 8;
  float* agg1 = p;  p += (size_t)N * 128;
  float* h2 = p;    p += (size_t)N * 16;
  float* aS2 = p;   p += (size_t)N;
  float* aD2 = p;   p += (size_t)N;
  float* m2 = p;    p += (size_t)N;
  float* s2 = p;    p += (size_t)N;
  float* agg2 = p;  p += (size_t)N * 16;

  const int T = 256;

  // --- init accumulators (every call: harness does not re-poison) ---
  fill_f32<<<cdiv((long long)N * 8, T), T, 0, stream>>>(m1, (long long)N * 8, -INFINITY);
  fill_f32<<<cdiv((long long)N * 8, T), T, 0, stream>>>(s1, (long long)N * 8, 0.f);
  fill_f32<<<cdiv((long long)N * 128, T), T, 0, stream>>>(agg1, (long long)N * 128, 0.f);
  fill_f32<<<cdiv(N, T), T, 0, stream>>>(m2, N, -INFINITY);
  fill_f32<<<cdiv(N, T), T, 0, stream>>>(s2, N, 0.f);
  fill_f32<<<cdiv((long long)N * 16, T), T, 0, stream>>>(agg2, (long long)N * 16, 0.f);

  // --- layer 1 ---
  // h1[N,128] = x[N,256] @ W1[256,128]  (WMMA fp32)
  gemm_f32_wmma<256, 128, 8><<<dim3(cdiv(N, 16), 1), 256, 0, stream>>>(x, W1, h1, N);
  node_att<8, 16><<<cdiv((long long)N * 8, T), T, 0, stream>>>(h1, attS1, attD1, aS1, aD1, N);
  edge_max<8><<<cdiv(ET * 8, T), T, 0, stream>>>(srcI, dstI, aS1, aD1, m1, E, N);
  edge_expsum<8><<<cdiv(ET * 8, T), T, 0, stream>>>(srcI, dstI, aS1, aD1, m1, s1, E, N);
  edge_agg<8, 16><<<cdiv(ET * 128, T), T, 0, stream>>>(srcI, dstI, aS1, aD1, m1, s1, h1, agg1, E, N);
  bias_elu<<<cdiv((long long)N * 128, T), T, 0, stream>>>(agg1, b1, (long long)N * 128, 128);

  // --- layer 2 ---
  // h2[N,16] = elu_out[N,128] @ W2[128,16]  (WMMA fp32)
  gemm_f32_wmma<128, 16, 2><<<dim3(cdiv(N, 16), 1), 64, 0, stream>>>(agg1, W2, h2, N);
  node_att<1, 16><<<cdiv(N, T), T, 0, stream>>>(h2, attS2, attD2, aS2, aD2, N);
  edge_max<1><<<cdiv(ET, T), T, 0, stream>>>(srcI, dstI, aS2, aD2, m2, E, N);
  edge_expsum<1><<<cdiv(ET, T), T, 0, stream>>>(srcI, dstI, aS2, aD2, m2, s2, E, N);
  edge_agg<1, 16><<<cdiv(ET * 16, T), T, 0, stream>>>(srcI, dstI, aS2, aD2, m2, s2, h2, agg2, E, N);

  // --- bias + log_softmax -> d_out ---
  bias_logsoftmax16<<<cdiv(N, T), T, 0, stream>>>(agg2, b2, out, N);
}